// Encoder_6725918785795
// MI455X (gfx1250) — compile-verified
//
#include <hip/hip_runtime.h>
#include <hip/hip_bf16.h>
#include <cstddef>

// ---------------------------------------------------------------------------
// 3-layer "bidirectional" GRU (both dirs scan forward), L=2048, B=24, H=75,
// IN=500.  Strategy:
//   K1 k_gx0    : GX0[t,d,b,0:225] = x[t,b]@w_ih0[d].T + b_ih0[d]
//                 big parallel GEMM (49152x500x450), V_WMMA_F32_16X16X4_F32,
//                 double-buffered b128 LDS staging.
//   K2 k_gru_l0 : 2 persistent WGs (fwd/bwd cell), 2048-step recurrence,
//                 gh = h@w_hh0.T via f32 WMMA, weights zero-padded in LDS so
//                 the K-loop is guard-free; GX0 slab staged to LDS with b128.
//   K3 k_gru_l12(l): layers 1 and 2 (two launches), input read from d_out
//                 slice written by the previous layer (stream-ordered dep).
// d_out layout: [L][B][6*75], cell c = 2*layer + dir owns columns c*75..c*75+74.
// ---------------------------------------------------------------------------

typedef __attribute__((ext_vector_type(2))) float v2f;
typedef __attribute__((ext_vector_type(4))) float v4f;
typedef __attribute__((ext_vector_type(8))) float v8f;

__device__ __forceinline__ v8f wmma_f32_k4(v2f a, v2f b, v8f c) {
  // D(16x16) = A(16x4) * B(4x16) + C   -- v_wmma_f32_16x16x4_f32
  return __builtin_amdgcn_wmma_f32_16x16x4_f32(
      /*neg_a=*/false, a, /*neg_b=*/false, b,
      /*c_mod=*/(short)0, c, /*reuse_a=*/false, /*reuse_b=*/false);
}

__device__ __forceinline__ float sigf(float x) {
  return 1.0f / (1.0f + __expf(-x));
}

// ---------------------------------------------------------------------------
// K1: GX0 precompute.  grid = (384, 30): blockIdx.x covers 8 m-tiles of 16
// rows (128 rows of the 49152-row A), blockIdx.y = d*15 + ntile.
// block = 256 threads = 8 waves, one 16x16 output tile per wave.
// K=500 = 15 full 32-chunks + 20 (=5 x float4) tail -> fully b128, no guards.
// ---------------------------------------------------------------------------
__global__ __launch_bounds__(256) void k_gx0(const float* __restrict__ x,
                                             const float* __restrict__ wih0,
                                             const float* __restrict__ bih0,
                                             float* __restrict__ gx0) {
  __shared__ float Bl[16][516];      // B tile (pitch 516: 16B-aligned, bank-spread)
  __shared__ float Al[2][128][36];   // double-buffered A chunk (pitch 36)

  const int tid = threadIdx.x;
  const int wave = tid >> 5, lane = tid & 31, ln = lane & 15, grp = lane >> 4;
  const int dn = blockIdx.y;
  const int d = dn / 15, nt = dn % 15;
  const int m0 = blockIdx.x * 128;

  // B tile once: rows n (16), cols k (512, pad >=500); b128 throughout.
  for (int idx = tid; idx < 16 * 128; idx += 256) {
    int r = idx >> 7, q = idx & 127;
    int ng = nt * 16 + r;
    v4f v = {0.f, 0.f, 0.f, 0.f};
    if (ng < 225 && q < 125)
      v = *(const v4f*)(wih0 + (size_t)(d * 225 + ng) * 500 + q * 4);
    *(v4f*)&Bl[r][q * 4] = v;
  }

  // C init = bias (lane's N column fixed)
  const int ng = nt * 16 + ln;
  const float bias = (ng < 225) ? bih0[d * 225 + ng] : 0.f;
  v8f acc;
#pragma unroll
  for (int r = 0; r < 8; r++) acc[r] = bias;

  // A chunk pipeline: load-to-regs early, wmma, then ds-store for next round.
  v4f areg[4];
  auto load_chunk = [&](int kc) {
    const int lim = (kc < 15) ? 8 : 5;  // chunk 15 covers k 480..499 only
#pragma unroll
    for (int i = 0; i < 4; i++) {
      int s = tid + 256 * i;
      int r = s >> 3, q = s & 7;
      v4f v = {0.f, 0.f, 0.f, 0.f};
      if (q < lim) v = *(const v4f*)(x + (size_t)(m0 + r) * 500 + kc * 32 + q * 4);
      areg[i] = v;
    }
  };
  auto store_chunk = [&](int buf) {
#pragma unroll
    for (int i = 0; i < 4; i++) {
      int s = tid + 256 * i;
      int r = s >> 3, q = s & 7;
      *(v4f*)&Al[buf][r][q * 4] = areg[i];
    }
  };

  load_chunk(0);
  store_chunk(0);

  const int mrow = wave * 16 + ln;
  for (int kc = 0; kc < 16; kc++) {
    __syncthreads();                    // buf kc&1 visible; buf (kc+1)&1 free
    if (kc + 1 < 16) load_chunk(kc + 1);  // global b128 in flight under wmma
    const int cur = kc & 1;
#pragma unroll
    for (int ks = 0; ks < 8; ks++) {
      const int kk = ks * 4 + grp * 2;
      v2f a = *(const v2f*)&Al[cur][mrow][kk];
      v2f b = *(const v2f*)&Bl[ln][kc * 32 + kk];
      acc = wmma_f32_k4(a, b, acc);
    }
    if (kc + 1 < 16) store_chunk((kc + 1) & 1);
  }

  // Scatter D tile: row m -> (t = m/24, b = m%24).  Half-wave writes 16
  // consecutive floats per row -> 64B segments.
  if (ng < 225) {
    const int mg_base = m0 + wave * 16 + grp * 8;
#pragma unroll
    for (int r = 0; r < 8; r++) {
      int m = mg_base + r;
      int t = m / 24, b = m % 24;
      gx0[((size_t)(t * 2 + d) * 24 + b) * 225 + ng] = acc[r];
    }
  }
}

// ---------------------------------------------------------------------------
// K2: layer-0 recurrence.  grid = 2 (dir), block = 960 (30 waves).
// Operands zero-padded to full tile extents -> guard-free WMMA K-loop.
// ---------------------------------------------------------------------------
__global__ __launch_bounds__(960) void k_gru_l0(const float* __restrict__ h0,
                                                const float* __restrict__ whh0,
                                                const float* __restrict__ bhh0,
                                                const float* __restrict__ gx0,
                                                float* __restrict__ out) {
  __shared__ float Wl[240][76];  // w_hh0[d][n][k], rows >=225 / col 75 zero
  __shared__ float Hl[32][76];   // hidden state, rows >=24 / col 75 zero
  __shared__ float Gh[24][228];  // gh pre-activations
  __shared__ float Gxl[5408];    // this step's GX0 slab (contiguous)
  __shared__ float Bh[240];      // b_hh, zero-padded

  const int tid = threadIdx.x;
  const int wave = tid >> 5, lane = tid & 31, ln = lane & 15, grp = lane >> 4;
  const int d = blockIdx.x;

  for (int idx = tid; idx < 240 * 76; idx += 960) {
    int n = idx / 76, k = idx % 76;
    Wl[n][k] = (n < 225 && k < 75) ? whh0[(size_t)(d * 225 + n) * 75 + k] : 0.f;
  }
  for (int idx = tid; idx < 240; idx += 960)
    Bh[idx] = (idx < 225) ? bhh0[d * 225 + idx] : 0.f;
  for (int idx = tid; idx < 32 * 76; idx += 960) {
    int m = idx / 76, k = idx % 76;
    Hl[m][k] = (m < 24 && k < 75) ? h0[(size_t)(d * 24 + m) * 75 + k] : 0.f;
  }
  __syncthreads();

  const int mt = wave & 1, nt = wave >> 1;  // 2 x 15 tiles over 30 waves
  const int nloc = nt * 16 + ln;
  const int mrow = mt * 16 + ln;
  const int mstore = mt * 16 + grp * 8;

  for (int t = 0; t < 2048; t++) {
    // Stage GX0 slab (24*225 floats, 16B-aligned) with b128 loads.
    const float* g = gx0 + (size_t)(t * 2 + d) * 5400;
    for (int idx = tid; idx < 1350; idx += 960)
      *(v4f*)&Gxl[idx * 4] = *(const v4f*)(g + idx * 4);

    v8f acc;
    const float bias = Bh[nloc];
#pragma unroll
    for (int r = 0; r < 8; r++) acc[r] = bias;
#pragma unroll
    for (int ks = 0; ks < 19; ks++) {   // guard-free: operands zero-padded
      const int k = ks * 4 + grp * 2;
      v2f a = *(const v2f*)&Hl[mrow][k];
      v2f b = *(const v2f*)&Wl[nloc][k];
      acc = wmma_f32_k4(a, b, acc);
    }
    if (nloc < 225) {
#pragma unroll
      for (int r = 0; r < 8; r++) {
        int m = mstore + r;
        if (m < 24) Gh[m][nloc] = acc[r];
      }
    }
    __syncthreads();

    for (int idx = tid; idx < 1800; idx += 960) {
      int m = idx / 75, j = idx % 75;
      const float* gm = &Gxl[m * 225];
      float r = sigf(gm[j] + Gh[m][j]);
      float z = sigf(gm[75 + j] + Gh[m][75 + j]);
      float nn = tanhf(gm[150 + j] + r * Gh[m][150 + j]);
      float hp = (1.f - z) * nn + z * Hl[m][j];
      Hl[m][j] = hp;
      out[((size_t)t * 24 + m) * 450 + d * 75 + j] = hp;
    }
    // Prefetch next step's GX0 slab (21.6 KB = 85 cachelines).
    if (t + 1 < 2048 && tid < 85) {
      const char* gn = (const char*)(g + 5400);
      __builtin_prefetch(gn + tid * 256, 0, 1);
    }
    __syncthreads();
  }
}

// ---------------------------------------------------------------------------
// K3: layers 1 & 2 (launched twice, l=1 then l=2).  grid = 2 (dir).
// Two-segment accumulation keeps xn and r*hn separate:
//   Gx over K=[0,150) (w_ih streamed from L2, row ptr clamped -> branchless),
//   Gh over K=[0,75)  (w_hh in LDS, zero-padded).
// ---------------------------------------------------------------------------
__global__ __launch_bounds__(960) void k_gru_l12(
    int l, const float* __restrict__ h0, const float* __restrict__ wih12,
    const float* __restrict__ whh12, const float* __restrict__ bih12,
    const float* __restrict__ bhh12, float* __restrict__ out) {
  __shared__ float Wh[240][76];   // w_hh, zero-padded
  __shared__ float Ab[32][228];   // [inp(0..151 pad) | h(152..227 pad)], rows>=24 zero
  __shared__ float Gx[24][228];
  __shared__ float Gh[24][228];
  __shared__ float Bi[240];
  __shared__ float Bh2[240];

  const int tid = threadIdx.x;
  const int wave = tid >> 5, lane = tid & 31, ln = lane & 15, grp = lane >> 4;
  const int d = blockIdx.x;
  const int li = (l - 1) * 2 + d;  // weight index
  const int cell = 2 * l + d;      // state / output slice index
  const float* __restrict__ wih = wih12 + (size_t)li * 225 * 150;

  for (int idx = tid; idx < 240 * 76; idx += 960) {
    int n = idx / 76, k = idx % 76;
    Wh[n][k] = (n < 225 && k < 75) ? whh12[(size_t)(li * 225 + n) * 75 + k] : 0.f;
  }
  for (int idx = tid; idx < 240; idx += 960) {
    Bi[idx] = (idx < 225) ? bih12[li * 225 + idx] : 0.f;
    Bh2[idx] = (idx < 225) ? bhh12[li * 225 + idx] : 0.f;
  }
  for (int idx = tid; idx < 32 * 228; idx += 960) {
    int m = idx / 228, k = idx % 228;
    float v = 0.f;
    if (m < 24 && k >= 152 && k < 227) {
      int kk = k - 152;
      v = (kk < 75) ? h0[(size_t)(cell * 24 + m) * 75 + kk] : 0.f;
    }
    Ab[m][k] = v;  // cols 0..149 overwritten each step; pads stay 0
  }
  __syncthreads();

  const int mt = wave & 1, nt = wave >> 1;
  const int nloc = nt * 16 + ln;
  const int mrow = mt * 16 + ln;
  const int mstore = mt * 16 + grp * 8;
  const int inoff = (l == 1) ? 0 : 150;
  // Clamped row pointer: branchless in-bounds loads; lanes nloc>=225 discarded
  // at the guarded store.
  const float* __restrict__ wrow = wih + (size_t)(nloc < 225 ? nloc : 224) * 150;

  for (int t = 0; t < 2048; t++) {
    // Stage this step's input rows (written by the previous layer's kernel).
    for (int idx = tid; idx < 24 * 75; idx += 960) {
      int m = idx / 75, q = idx % 75;
      *(v2f*)&Ab[m][q * 2] =
          *(const v2f*)(out + ((size_t)t * 24 + m) * 450 + inoff + q * 2);
    }
    __syncthreads();

    // ---- segment 1: Gx = inp @ w_ih.T + b_ih  (K=0..147 unguarded + peel)
    v8f acc;
    {
      const float bias = Bi[nloc];
#pragma unroll
      for (int r = 0; r < 8; r++) acc[r] = bias;
    }
#pragma unroll 2
    for (int ks = 0; ks < 37; ks++) {   // k <= 146: fully in-bounds
      const int k = ks * 4 + grp * 2;
      v2f a = *(const v2f*)&Ab[mrow][k];
      v2f b = *(const v2f*)(wrow + k);
      acc = wmma_f32_k4(a, b, acc);
    }
    {  // peeled partial step: k = 148 (grp0, valid) / 150 (grp1, zero)
      const int k = 148 + grp * 2;
      v2f a = *(const v2f*)&Ab[mrow][k];  // cols 150,151 are zero
      v2f b = {0.f, 0.f};
      if (grp == 0) b = *(const v2f*)(wrow + 148);
      acc = wmma_f32_k4(a, b, acc);
    }
    if (nloc < 225) {
#pragma unroll
      for (int r = 0; r < 8; r++) {
        int m = mstore + r;
        if (m < 24) Gx[m][nloc] = acc[r];
      }
    }

    // ---- segment 2: Gh = h @ w_hh.T + b_hh  (K = 152..227, guard-free)
    {
      const float bias = Bh2[nloc];
#pragma unroll
      for (int r = 0; r < 8; r++) acc[r] = bias;
    }
#pragma unroll
    for (int ks = 0; ks < 19; ks++) {
      const int k = 152 + ks * 4 + grp * 2;
      v2f a = *(const v2f*)&Ab[mrow][k];
      v2f b = *(const v2f*)&Wh[nloc][k - 152];
      acc = wmma_f32_k4(a, b, acc);
    }
    if (nloc < 225) {
#pragma unroll
      for (int r = 0; r < 8; r++) {
        int m = mstore + r;
        if (m < 24) Gh[m][nloc] = acc[r];
      }
    }
    __syncthreads();

    // ---- gates + state update
    for (int idx = tid; idx < 1800; idx += 960) {
      int m = idx / 75, j = idx % 75;
      float r = sigf(Gx[m][j] + Gh[m][j]);
      float z = sigf(Gx[m][75 + j] + Gh[m][75 + j]);
      float nn = tanhf(Gx[m][150 + j] + r * Gh[m][150 + j]);
      float hp = (1.f - z) * nn + z * Ab[m][152 + j];
      Ab[m][152 + j] = hp;
      out[((size_t)t * 24 + m) * 450 + cell * 75 + j] = hp;
    }
    // Prefetch next step's input rows (24 rows x 600 B, strided by 1800 B).
    if (t + 1 < 2048 && tid < 72) {
      int m = tid / 3, c = tid % 3;
      const char* p =
          (const char*)(out + ((size_t)(t + 1) * 24 + m) * 450 + inoff);
      __builtin_prefetch(p + c * 256, 0, 1);
    }
    __syncthreads();
  }
}

// ---------------------------------------------------------------------------
extern "C" void kernel_launch(void* const* d_in, const int* in_sizes, int n_in,
                              void* d_out, int out_size, void* d_ws,
                              size_t ws_size, hipStream_t stream) {
  (void)in_sizes; (void)n_in; (void)out_size; (void)ws_size;
  const float* x     = (const float*)d_in[0];
  const float* h0    = (const float*)d_in[1];
  const float* wih0  = (const float*)d_in[2];
  const float* whh0  = (const float*)d_in[3];
  const float* bih0  = (const float*)d_in[4];
  const float* bhh0  = (const float*)d_in[5];
  const float* wih12 = (const float*)d_in[6];
  const float* whh12 = (const float*)d_in[7];
  const float* bih12 = (const float*)d_in[8];
  const float* bhh12 = (const float*)d_in[9];
  float* out = (float*)d_out;
  float* gx0 = (float*)d_ws;  // 2048*2*24*225 f32 = 88.5 MB scratch

  dim3 g1(384, 30);  // 3072 m-tiles / 8 per block, 2 dirs * 15 n-tiles
  k_gx0<<<g1, 256, 0, stream>>>(x, wih0, bih0, gx0);
  k_gru_l0<<<2, 960, 0, stream>>>(h0, whh0, bhh0, gx0, out);
  k_gru_l12<<<2, 960, 0, stream>>>(1, h0, wih12, whh12, bih12, bhh12, out);
  k_gru_l12<<<2, 960, 0, stream>>>(2, h0, wih12, whh12, bih12, bhh12, out);
}